// Sampler_3006477107624
// MI455X (gfx1250) — compile-verified
//
#include <hip/hip_runtime.h>
#include <stdint.h>

#define BLOCK     1024
#define VOCAB     128000
#define NV4       (VOCAB / 4)          // 32000 float4 per row
#define HBUCKETS  4096                 // 12-bit key histogram
#define CAND_MAX  4096
#define TILE4     BLOCK                // float4 per staged tile
#define NTILES    ((NV4 + TILE4 - 1) / TILE4)   // 32 (last one partial)

#if defined(__has_builtin)
#  if __has_builtin(__builtin_amdgcn_global_load_async_to_lds_b128)
#    define HAVE_ASYNC_LDS 1
#  endif
#endif

typedef int v4i __attribute__((ext_vector_type(4)));
typedef __attribute__((address_space(1))) v4i* gv4i_p;   // global
typedef __attribute__((address_space(3))) v4i* lv4i_p;   // LDS

// Monotone uint32 transform of float (total order), and its inverse.
__device__ __forceinline__ unsigned sortkey(float f) {
    unsigned x = __float_as_uint(f);
    return (x & 0x80000000u) ? ~x : (x | 0x80000000u);
}
__device__ __forceinline__ float sortkey_inv(unsigned k) {
    unsigned x = (k & 0x80000000u) ? (k & 0x7fffffffu) : ~k;
    return __uint_as_float(x);
}

__global__ __launch_bounds__(BLOCK) void sampler_kernel(
    const float* __restrict__ logits,
    const float* __restrict__ temps,
    const float* __restrict__ min_ps,
    const float* __restrict__ top_ps,
    const int*   __restrict__ top_ks,
    const float* __restrict__ uvec,
    int*         __restrict__ out)
{
    __shared__ unsigned            hist[HBUCKETS];     // 16 KB
    __shared__ unsigned long long  cand[CAND_MAX];     // 32 KB
    __shared__ float               scanA[BLOCK];       // 4 KB
    __shared__ float               scanB[BLOCK];       // 4 KB
    __shared__ float               pArr[BLOCK];        // 4 KB
    __shared__ unsigned            upart[BLOCK];       // 4 KB
    __shared__ __align__(16) float stage[2][TILE4 * 4];// 32 KB (async staging)
    __shared__ unsigned            candCount;
    __shared__ int                 threshBucket;
    __shared__ int                 sampledIdx;
    __shared__ float               s_m, s_Z;

    const int b = blockIdx.x;
    const int t = threadIdx.x;
    const float*  row  = logits + (size_t)b * VOCAB;
    const float4* row4 = reinterpret_cast<const float4*>(row);

    const float invT = 1.0f / temps[b];
    int K = top_ks[b];
    if (K < 1) K = 1;
    if (K > BLOCK) K = BLOCK;
    const float topp = top_ps[b];
    const float minp = min_ps[b];
    const float u    = uvec[b];

    // ---- init shared ----
    for (int i = t; i < HBUCKETS; i += BLOCK) hist[i] = 0u;
    if (t == 0) { candCount = 0u; threshBucket = 0; sampledIdx = 0x7fffffff; }
    __syncthreads();

    // =====================================================================
    // Pass 1: streaming max + rescaled softmax sum + 12-bit key histogram
    // =====================================================================
    float m_loc = -3.402823466e38f, s_loc = 0.0f;
    for (int c = 0; c < NTILES; ++c) {
        const int i4 = t + c * BLOCK;
        if (i4 < NV4) {
            const int nf = i4 + BLOCK;
            if (nf < NV4) __builtin_prefetch(row4 + nf, 0, 1);   // global_prefetch_b8
            const float4 v = row4[i4];
            const float vv[4] = {v.x, v.y, v.z, v.w};
#pragma unroll
            for (int k = 0; k < 4; ++k) {
                const float z = vv[k] * invT;
                if (z > m_loc) { s_loc = s_loc * __expf(m_loc - z) + 1.0f; m_loc = z; }
                else           { s_loc += __expf(z - m_loc); }
                atomicAdd(&hist[sortkey(vv[k]) >> 20], 1u);
            }
        }
    }
    // block reduce (m, s) with rescaling
    scanA[t] = m_loc; scanB[t] = s_loc;
    __syncthreads();
    for (int s = BLOCK / 2; s > 0; s >>= 1) {
        if (t < s) {
            const float m1 = scanA[t], m2 = scanA[t + s];
            const float s1 = scanB[t], s2 = scanB[t + s];
            const float M = fmaxf(m1, m2);
            scanB[t] = s1 * __expf(m1 - M) + s2 * __expf(m2 - M);
            scanA[t] = M;
        }
        __syncthreads();
    }
    if (t == 0) { s_m = scanA[0]; s_Z = scanB[0]; }
    __syncthreads();
    const float m = s_m, Z = s_Z;
    __syncthreads();   // scanA/scanB free for reuse

    // =====================================================================
    // Histogram prefix scan -> bucket containing the K-th largest key
    // =====================================================================
    const unsigned c0 = hist[4 * t + 0], c1 = hist[4 * t + 1];
    const unsigned c2 = hist[4 * t + 2], c3 = hist[4 * t + 3];
    upart[t] = c0 + c1 + c2 + c3;
    __syncthreads();
    for (int off = 1; off < BLOCK; off <<= 1) {       // inclusive scan (u32)
        const unsigned v = upart[t];
        const unsigned a = (t >= off) ? upart[t - off] : 0u;
        __syncthreads();
        upart[t] = v + a;
        __syncthreads();
    }
    {
        const unsigned grpExc = upart[t] - (c0 + c1 + c2 + c3);
        const unsigned pin[4] = {grpExc + c0, grpExc + c0 + c1,
                                 grpExc + c0 + c1 + c2, grpExc + c0 + c1 + c2 + c3};
        const unsigned cc[4]  = {c0, c1, c2, c3};
#pragma unroll
        for (int k = 0; k < 4; ++k) {
            const unsigned above = (unsigned)VOCAB - pin[k]; // count in buckets > b
            const unsigned withb = above + cc[k];            // count in buckets >= b
            if (withb >= (unsigned)K && above < (unsigned)K) threshBucket = 4 * t + k;
        }
    }
    __syncthreads();
    const unsigned thrKey = ((unsigned)threshBucket) << 20;

    // =====================================================================
    // Pass 2: collect candidates (key >= thrKey) via async global->LDS tiles
    // =====================================================================
#if defined(HAVE_ASYNC_LDS)
    auto issueTile = [&](int tile, int buf) {
        const int base4 = tile * TILE4 + t;
        if (base4 < NV4) {
            gv4i_p g = (gv4i_p)(void*)(row4 + base4);
            lv4i_p l = (lv4i_p)(&stage[buf][4 * t]);
            __builtin_amdgcn_global_load_async_to_lds_b128(g, l, 0, 0);
        }
    };
    issueTile(0, 0);
    for (int tile = 0; tile < NTILES; ++tile) {
        const int buf = tile & 1;
        if (tile + 1 < NTILES) issueTile(tile + 1, buf ^ 1);
        // last tile is partial (some waves skip the prefetch) -> wait 0 there
#  if __has_builtin(__builtin_amdgcn_s_wait_asynccnt)
        if (tile + 2 < NTILES) __builtin_amdgcn_s_wait_asynccnt(1);
        else                   __builtin_amdgcn_s_wait_asynccnt(0);
#  else
        asm volatile("s_wait_asynccnt 0" ::: "memory");
#  endif
        const int base4 = tile * TILE4 + t;
        if (base4 < NV4) {
            const float4 v = *reinterpret_cast<const float4*>(&stage[buf][4 * t]);
            const float vv[4] = {v.x, v.y, v.z, v.w};
            const int ebase = base4 * 4;
#pragma unroll
            for (int k = 0; k < 4; ++k) {
                const unsigned key = sortkey(vv[k]);
                if (key >= thrKey) {
                    const unsigned pos = atomicAdd(&candCount, 1u);
                    if (pos < CAND_MAX)
                        cand[pos] = ((unsigned long long)key << 32) |
                                    (unsigned)(~(unsigned)(ebase + k));
                }
            }
        }
    }
#else
    for (int c = 0; c < NTILES; ++c) {
        const int i4 = t + c * BLOCK;
        if (i4 < NV4) {
            const float4 v = row4[i4];
            const float vv[4] = {v.x, v.y, v.z, v.w};
            const int ebase = i4 * 4;
#pragma unroll
            for (int k = 0; k < 4; ++k) {
                const unsigned key = sortkey(vv[k]);
                if (key >= thrKey) {
                    const unsigned pos = atomicAdd(&candCount, 1u);
                    if (pos < CAND_MAX)
                        cand[pos] = ((unsigned long long)key << 32) |
                                    (unsigned)(~(unsigned)(ebase + k));
                }
            }
        }
    }
#endif
    __syncthreads();

    unsigned C = candCount;
    if (C > CAND_MAX) C = CAND_MAX;

    // pad to power of two (key 0 sorts to the tail of a descending sort)
    int Npad = 2;
    while (Npad < (int)C) Npad <<= 1;
    for (int i = t; i < Npad; i += BLOCK)
        if (i >= (int)C) cand[i] = 0ull;
    __syncthreads();

    // ---- bitonic sort, descending, 64-bit (key | ~index) ----
    for (int k = 2; k <= Npad; k <<= 1) {
        for (int j = k >> 1; j > 0; j >>= 1) {
            for (int i = t; i < Npad; i += BLOCK) {
                const int ixj = i ^ j;
                if (ixj > i) {
                    const unsigned long long a = cand[i], bb = cand[ixj];
                    const bool descBlk = ((i & k) == 0);
                    if (descBlk ? (a < bb) : (a > bb)) { cand[i] = bb; cand[ixj] = a; }
                }
            }
            __syncthreads();
        }
    }

    // =====================================================================
    // Probs for top ranks, scans, top-p / min-p filters, inverse-CDF draw
    // =====================================================================
    const int Kn = ((int)C < K) ? (int)C : K;
    {
        float p = 0.0f;
        if (t < Kn) {
            const unsigned key = (unsigned)(cand[t] >> 32);
            const float v = sortkey_inv(key);
            p = __expf(v * invT - m) / Z;
        }
        pArr[t] = p;
        scanA[t] = p;
    }
    __syncthreads();
    for (int off = 1; off < BLOCK; off <<= 1) {       // inclusive scan #1
        const float v = scanA[t];
        const float a = (t >= off) ? scanA[t - off] : 0.0f;
        __syncthreads();
        scanA[t] = v + a;
        __syncthreads();
    }
    {
        const float p      = pArr[t];
        const float before = scanA[t] - p;            // prefix mass before rank t
        const float thr    = minp / Z;                // min_p * p_max, p_max = 1/Z
        const float p1     = (t < Kn && before <= topp) ? p : 0.0f;
        const float p2     = (p1 >= thr) ? p1 : 0.0f;
        scanB[t] = p2;
    }
    __syncthreads();
    for (int off = 1; off < BLOCK; off <<= 1) {       // inclusive scan #2 (filtered CDF)
        const float v = scanB[t];
        const float a = (t >= off) ? scanB[t - off] : 0.0f;
        __syncthreads();
        scanB[t] = v + a;
        __syncthreads();
    }
    {
        const float total  = scanB[BLOCK - 1];
        const float target = u * total;
        const float cdf    = scanB[t];
        const float prev   = (t == 0) ? 0.0f : scanB[t - 1];
        if (cdf > target && prev <= target) atomicMin(&sampledIdx, t);
    }
    __syncthreads();
    if (t == 0) {
        int si = sampledIdx;
        if (si == 0x7fffffff) si = 0;                 // matches argmax-of-all-false -> 0
        const unsigned idx = ~(unsigned)(cand[si] & 0xffffffffu);
        out[b] = (int)idx;
    }
}

extern "C" void kernel_launch(void* const* d_in, const int* in_sizes, int n_in,
                              void* d_out, int out_size, void* d_ws, size_t ws_size,
                              hipStream_t stream) {
    (void)n_in; (void)d_ws; (void)ws_size;
    const float* logits = (const float*)d_in[0];
    const float* temps  = (const float*)d_in[1];
    const float* minps  = (const float*)d_in[2];
    const float* topps  = (const float*)d_in[3];
    const int*   topks  = (const int*)d_in[4];
    const float* u      = (const float*)d_in[5];
    const int B = in_sizes[1];                        // 128 rows
    (void)out_size;
    sampler_kernel<<<B, BLOCK, 0, stream>>>(logits, temps, minps, topps, topks, u,
                                            (int*)d_out);
}